// HydraMultivariate_53206054863452
// MI455X (gfx1250) — compile-verified
//
#include <hip/hip_runtime.h>

typedef __attribute__((ext_vector_type(2))) float v2f;
typedef __attribute__((ext_vector_type(8))) float v8f;

#define HY_B    128
#define HY_C    12
#define HY_L    2048
#define HY_K    8
#define HY_G2   32     // _G
#define HY_NCP  6
#define HY_ND   8
#define HY_OPB  8192   // outputs per batch row
#define HY_PAD  512    // max halo = 4*d, d<=128
#define HY_PADL (HY_L + 2 * HY_PAD)   // 3072 floats = 12 KB LDS

template <int DI, int DIF>
__global__ __launch_bounds__(32)
void hydra_wmma_kernel(const float* __restrict__ X,
                       const float* __restrict__ W,
                       const int*   __restrict__ I,
                       float*       __restrict__ out) {
    constexpr int comb = DI * 2 + DIF;
    constexpr int d    = 1 << DI;          // dilation (compile-time)
    constexpr int Lp   = HY_L - DIF;       // input length == output length

    __shared__ alignas(16) float sPad[HY_PADL];

    const int lane = threadIdx.x;          // wave32, one wave per workgroup
    const int g    = blockIdx.x;           // group 0..31
    const int b    = blockIdx.z;           // batch 0..127

    // ---- zero the halo margins; interior is overwritten below. DS ops of a
    // single wave complete in order, so no barrier is needed in between. ----
    const float4 z4 = make_float4(0.f, 0.f, 0.f, 0.f);
#pragma unroll
    for (int l = lane * 4; l < HY_PAD; l += 128)
        *(float4*)(sPad + l) = z4;
    for (int l = (HY_PAD + HY_L - 16) + lane * 4; l < HY_PADL; l += 128)
        *(float4*)(sPad + l) = z4;

    // ---- gather 6-channel subset sum into padded LDS, float4-vectorized ----
    // diff commutes with channel summation: diff(sum_c X_c) == sum_c diff(X_c)
    const int* Ig = I + (comb * HY_G2 + g) * HY_NCP;
    const float* Xb = X + (size_t)b * HY_C * HY_L;
    const float* p0 = Xb + (size_t)Ig[0] * HY_L;
    const float* p1 = Xb + (size_t)Ig[1] * HY_L;
    const float* p2 = Xb + (size_t)Ig[2] * HY_L;
    const float* p3 = Xb + (size_t)Ig[3] * HY_L;
    const float* p4 = Xb + (size_t)Ig[4] * HY_L;
    const float* p5 = Xb + (size_t)Ig[5] * HY_L;

    for (int l = lane * 4; l < HY_L; l += 128) {
        float4 a0 = *(const float4*)(p0 + l);
        float4 a1 = *(const float4*)(p1 + l);
        float4 a2 = *(const float4*)(p2 + l);
        float4 a3 = *(const float4*)(p3 + l);
        float4 a4 = *(const float4*)(p4 + l);
        float4 a5 = *(const float4*)(p5 + l);
        float sx = a0.x + a1.x + a2.x + a3.x + a4.x + a5.x;
        float sy = a0.y + a1.y + a2.y + a3.y + a4.y + a5.y;
        float sz = a0.z + a1.z + a2.z + a3.z + a4.z + a5.z;
        float sw = a0.w + a1.w + a2.w + a3.w + a4.w + a5.w;
        if (DIF == 0) {
            *(float4*)(sPad + HY_PAD + l) = make_float4(sx, sy, sz, sw);
        } else {
            float t = 0.0f;
            if (l + 4 < HY_L)
                t = p0[l+4] + p1[l+4] + p2[l+4] + p3[l+4] + p4[l+4] + p5[l+4];
            const float rw = (l + 4 < HY_L) ? (t - sw) : 0.0f;  // slot Lp stays 0
            *(float4*)(sPad + HY_PAD + l) = make_float4(sy - sx, sz - sy, sw - sz, rw);
        }
    }
    __syncthreads();

    // ---- WMMA operand roles (V_WMMA_F32_16X16X4_F32) ----
    // A 16x4: lane supplies row M = lane&15 for K rows {0,1} (lanes 0-15) or
    // {2,3} (lanes 16-31). Rows 8-15 carry NEGATED kernel weights, so D rows
    // 8-15 (lanes 16-31 of D) hold -Zc: max-scan there == argmin of Zc.
    const int   nLow = lane & 15;
    const int   hi   = lane >> 4;          // 0: max half, 1: min half of D
    const int   k    = lane & 7;           // kernel index
    const int   tb   = hi * 2;             // first tap this lane supplies per K-chunk
    const float sgn  = (lane & 8) ? -1.0f : 1.0f;   // negate A rows 8-15

    const float* Wg = W + ((size_t)comb * (HY_G2 * HY_K) + g * HY_K + k) * 9;
    float wk[9];
#pragma unroll
    for (int t = 0; t < 9; ++t) wk[t] = Wg[t] * sgn;

    v2f A0 = { wk[tb + 0], wk[tb + 1] };                  // taps 0..3
    v2f A1 = { wk[tb + 4], wk[tb + 5] };                  // taps 4..7
    v2f A2 = { (hi == 0) ? wk[8] : 0.0f, 0.0f };          // tap 8 (+3 zero K slots)

    float accum[8];                        // lanes<16: sum(max_v); lanes>=16: counts
#pragma unroll
    for (int r = 0; r < 8; ++r) accum[r] = 0.0f;

    // per-lane invariant base for tap 'tb' of column nLow (padded coordinates);
    // all 5 tap offsets {0, d, 4d, 5d, 8d} are compile-time immediates.
    const int off0 = HY_PAD + nLow + (tb - 4) * d;

    // ---- 128 column tiles of 16, 3 chained f32 WMMAs each ----
    for (int l0 = 0; l0 < HY_L; l0 += 16) {
        const int base = off0 + l0;
        v2f B0 = { sPad[base],           sPad[base + d] };
        v2f B1 = { sPad[base + 4 * d],   sPad[base + 5 * d] };
        const int a8 = base + ((hi == 0) ? 8 * d : 0);    // keep address in range
        const float t8 = sPad[a8];
        v2f B2 = { (hi == 0) ? t8 : 0.0f, 0.0f };

        v8f z = {0.f,0.f,0.f,0.f,0.f,0.f,0.f,0.f};
        z = __builtin_amdgcn_wmma_f32_16x16x4_f32(false, A0, false, B0,
                                                  (short)0, z, false, false);
        z = __builtin_amdgcn_wmma_f32_16x16x4_f32(false, A1, false, B1,
                                                  (short)0, z, false, false);
        z = __builtin_amdgcn_wmma_f32_16x16x4_f32(false, A2, false, B2,
                                                  (short)0, z, false, false);

        // first-occurrence max tree scan (depth 3). Strict '>' at every node
        // keeps the leftmost index among ties, matching jnp.argmax / argmin.
        float m01 = fmaxf(z[0], z[1]); int i01 = (z[1] > z[0]) ? 1 : 0;
        float m23 = fmaxf(z[2], z[3]); int i23 = (z[3] > z[2]) ? 3 : 2;
        float m45 = fmaxf(z[4], z[5]); int i45 = (z[5] > z[4]) ? 5 : 4;
        float m67 = fmaxf(z[6], z[7]); int i67 = (z[7] > z[6]) ? 7 : 6;
        float m03 = fmaxf(m01, m23);   int i03 = (m23 > m01) ? i23 : i01;
        float m47 = fmaxf(m45, m67);   int i47 = (m67 > m45) ? i67 : i45;
        float best = fmaxf(m03, m47);  int bi  = (m47 > m03) ? i47 : i03;

        const bool ok = (DIF == 0) || ((l0 + nLow) < Lp);  // Lout == Lp
        float contrib = (hi != 0) ? 1.0f : best;           // count vs max-value
        contrib = ok ? contrib : 0.0f;
#pragma unroll
        for (int r = 0; r < 8; ++r)
            accum[r] += (bi == r) ? contrib : 0.0f;
    }

    // ---- two independent 16-lane butterfly reductions (halves don't mix) ----
#pragma unroll
    for (int off = 1; off <= 8; off <<= 1) {
#pragma unroll
        for (int r = 0; r < 8; ++r)
            accum[r] += __shfl_xor(accum[r], off, 32);
    }

    float* obase = out + (size_t)b * HY_OPB + (size_t)comb * 512 + g * HY_K;
    if (lane == 0) {       // count_max block
        *(float4*)(obase + 0) = make_float4(accum[0], accum[1], accum[2], accum[3]);
        *(float4*)(obase + 4) = make_float4(accum[4], accum[5], accum[6], accum[7]);
    }
    if (lane == 16) {      // count_min block
        *(float4*)(obase + 256) = make_float4(accum[0], accum[1], accum[2], accum[3]);
        *(float4*)(obase + 260) = make_float4(accum[4], accum[5], accum[6], accum[7]);
    }
}

extern "C" void kernel_launch(void* const* d_in, const int* in_sizes, int n_in,
                              void* d_out, int out_size, void* d_ws, size_t ws_size,
                              hipStream_t stream) {
    const float* X = (const float*)d_in[0];   // [128,12,2048] f32
    const float* W = (const float*)d_in[1];   // [8,2,256,1,9] f32 (pre-normalized)
    const int*   I = (const int*)d_in[2];     // [8,2,32,6] i32
    float* out = (float*)d_out;               // [128,8192] f32

    dim3 grid(HY_G2, 1, HY_B);                // (g, -, b), one launch per (di,diff)
#define HY_LAUNCH(DIv, DIFv) \
    hydra_wmma_kernel<DIv, DIFv><<<grid, 32, 0, stream>>>(X, W, I, out)
    HY_LAUNCH(0, 0); HY_LAUNCH(0, 1);
    HY_LAUNCH(1, 0); HY_LAUNCH(1, 1);
    HY_LAUNCH(2, 0); HY_LAUNCH(2, 1);
    HY_LAUNCH(3, 0); HY_LAUNCH(3, 1);
    HY_LAUNCH(4, 0); HY_LAUNCH(4, 1);
    HY_LAUNCH(5, 0); HY_LAUNCH(5, 1);
    HY_LAUNCH(6, 0); HY_LAUNCH(6, 1);
    HY_LAUNCH(7, 0); HY_LAUNCH(7, 1);
#undef HY_LAUNCH
}